// Loss_6150393168105
// MI455X (gfx1250) — compile-verified
//
#include <hip/hip_runtime.h>
#include <math.h>
#include <stdint.h>

// ---------------------------------------------------------------------------
// YOLO-v1 style loss, fully bandwidth-bound streaming reduction.
// pred,tgt: fp32 [N,7,7,30]  ->  5 fp32 scalars.
// Strategy: async global->LDS staging (CDNA5 ASYNCcnt path) of 32 cells/wave,
// per-lane cell processing from LDS, wave32 shuffle + LDS + f64 atomic reduce.
// ---------------------------------------------------------------------------

#define WAVES        8
#define BLOCK        256
#define CELL_F       30                      // floats per cell
#define CHUNK_CELLS  32                      // cells per wave-chunk
#define CHUNK_FLOATS (CHUNK_CELLS * CELL_F)  // 960 floats
#define CHUNK_BYTES  (CHUNK_FLOATS * 4)      // 3840 bytes

__device__ __forceinline__ void process_cell(const float p[CELL_F],
                                             const float t[CELL_F],
                                             float acc[5]) {
  const float cell = 1.0f / 7.0f;

  // target box 0 -> LTRB (mirror reference expression structure)
  const float tx1 = t[0] * cell - t[2] * 0.5f;
  const float ty1 = t[1] * cell - t[3] * 0.5f;
  const float tx2 = t[0] * cell + t[2] * 0.5f;
  const float ty2 = t[1] * cell + t[3] * 0.5f;
  const float area_t = (tx2 - tx1) * (ty2 - ty1);

  float iou0, iou1;
  {
    const float px1 = p[0] * cell - p[2] * 0.5f;
    const float py1 = p[1] * cell - p[3] * 0.5f;
    const float px2 = p[0] * cell + p[2] * 0.5f;
    const float py2 = p[1] * cell + p[3] * 0.5f;
    const float w = fmaxf(fminf(px2, tx2) - fmaxf(px1, tx1), 0.0f);
    const float h = fmaxf(fminf(py2, ty2) - fmaxf(py1, ty1), 0.0f);
    const float inter = w * h;
    const float area_p = (px2 - px1) * (py2 - py1);
    iou0 = inter / (area_p + area_t - inter);
  }
  {
    const float px1 = p[5] * cell - p[7] * 0.5f;
    const float py1 = p[6] * cell - p[8] * 0.5f;
    const float px2 = p[5] * cell + p[7] * 0.5f;
    const float py2 = p[6] * cell + p[8] * 0.5f;
    const float w = fmaxf(fminf(px2, tx2) - fmaxf(px1, tx1), 0.0f);
    const float h = fmaxf(fminf(py2, ty2) - fmaxf(py1, ty1), 0.0f);
    const float inter = w * h;
    const float area_p = (px2 - px1) * (py2 - py1);
    iou1 = inter / (area_p + area_t - inter);
  }

  const bool  b0    = (iou0 >= iou1);           // jnp.argmax: first max wins
  const float riou  = fmaxf(iou0, iou1);
  const float obj   = (t[4] == 1.0f) ? 1.0f : 0.0f;
  const float noobj = (t[4] == 0.0f) ? 1.0f : 0.0f;

  // select responsible box (cndmask, no dynamic register indexing)
  const float rtx = b0 ? t[0] : t[5];
  const float rty = b0 ? t[1] : t[6];
  const float rtw = b0 ? t[2] : t[7];
  const float rth = b0 ? t[3] : t[8];
  const float rpx = b0 ? p[0] : p[5];
  const float rpy = b0 ? p[1] : p[6];
  const float rpw = b0 ? p[2] : p[7];
  const float rph = b0 ? p[3] : p[8];
  const float rpc = b0 ? p[4] : p[9];

  const float dx = rtx - rpx, dy = rty - rpy;
  acc[0] += obj * (dx * dx + dy * dy);

  const float dw = sqrtf(rtw) - sqrtf(rpw);
  const float dh = sqrtf(rth) - sqrtf(rph);
  acc[1] += obj * (dw * dw + dh * dh);

  const float dob = riou - rpc;
  acc[2] += obj * dob * dob;

  const float d4 = t[4] - p[4];
  const float d9 = t[9] - p[9];
  acc[3] += noobj * (d4 * d4 + d9 * d9);

  float cs = 0.0f;
#pragma unroll
  for (int c = 10; c < 30; ++c) {
    const float d = t[c] - p[c];
    cs += d * d;
  }
  acc[4] += obj * cs;
}

__global__ void __launch_bounds__(BLOCK)
yolo_loss_kernel(const float* __restrict__ pred, const float* __restrict__ tgt,
                 double* __restrict__ ws, int nchunks, int rem_cells,
                 long long nfull_cells) {
  __shared__ float smem[WAVES * 2 * CHUNK_FLOATS];  // per wave: [pred 960 | tgt 960]
  __shared__ float red[WAVES][5];

  const int lane = threadIdx.x & 31;
  const int wave = threadIdx.x >> 5;

  // low 32 bits of the generic pointer to a __shared__ object == LDS byte offset
  const unsigned smem_base = (unsigned)(uintptr_t)&smem[0];
  const unsigned lds_p = smem_base + (unsigned)wave * (2u * CHUNK_BYTES);
  const unsigned lds_t = lds_p + CHUNK_BYTES;

  float acc[5] = {0.f, 0.f, 0.f, 0.f, 0.f};

  const int gw = blockIdx.x * WAVES + wave;
  const int gstride = gridDim.x * WAVES;

  for (int ch = gw; ch < nchunks; ch += gstride) {
    const unsigned long long gp =
        (unsigned long long)(uintptr_t)pred + (unsigned long long)ch * CHUNK_BYTES;
    const unsigned long long gt =
        (unsigned long long)(uintptr_t)tgt + (unsigned long long)ch * CHUNK_BYTES;

    // ensure prior-chunk ds_loads have drained before overwriting the buffer
    asm volatile("s_wait_dscnt 0" ::: "memory");

    // 32 cells * 120B = 3840B per tensor: 7 x b128 (512B/instr) + 1 x b64 (256B)
#pragma unroll
    for (int k = 0; k < 7; ++k) {
      asm volatile("global_load_async_to_lds_b128 %0, %1, off"
                   :: "v"(lds_p + (unsigned)k * 512u + (unsigned)lane * 16u),
                      "v"(gp + (unsigned)k * 512u + (unsigned)lane * 16u)
                   : "memory");
    }
    asm volatile("global_load_async_to_lds_b64 %0, %1, off"
                 :: "v"(lds_p + 3584u + (unsigned)lane * 8u),
                    "v"(gp + 3584u + (unsigned)lane * 8u)
                 : "memory");
#pragma unroll
    for (int k = 0; k < 7; ++k) {
      asm volatile("global_load_async_to_lds_b128 %0, %1, off"
                   :: "v"(lds_t + (unsigned)k * 512u + (unsigned)lane * 16u),
                      "v"(gt + (unsigned)k * 512u + (unsigned)lane * 16u)
                   : "memory");
    }
    asm volatile("global_load_async_to_lds_b64 %0, %1, off"
                 :: "v"(lds_t + 3584u + (unsigned)lane * 8u),
                    "v"(gt + 3584u + (unsigned)lane * 8u)
                 : "memory");

    asm volatile("s_wait_asynccnt 0" ::: "memory");

    // each lane owns one cell; stride 30 dwords -> conflict-free b64 ds reads
    const float* cp = &smem[wave * (2 * CHUNK_FLOATS) + lane * CELL_F];
    const float* ct = cp + CHUNK_FLOATS;
    float p[CELL_F], t[CELL_F];
#pragma unroll
    for (int i = 0; i < CELL_F; ++i) p[i] = cp[i];
#pragma unroll
    for (int i = 0; i < CELL_F; ++i) t[i] = ct[i];

    process_cell(p, t, acc);
  }

  // remainder cells (cells % 32) — N*49 is divisible by 32 here, but stay general
  if (blockIdx.x == 0 && wave == 0 && lane < rem_cells) {
    const float* cp = pred + (nfull_cells + lane) * CELL_F;
    const float* ct = tgt + (nfull_cells + lane) * CELL_F;
    float p[CELL_F], t[CELL_F];
#pragma unroll
    for (int i = 0; i < CELL_F; ++i) p[i] = cp[i];
#pragma unroll
    for (int i = 0; i < CELL_F; ++i) t[i] = ct[i];
    process_cell(p, t, acc);
  }

  // wave32 tree reduction
#pragma unroll
  for (int i = 0; i < 5; ++i) {
#pragma unroll
    for (int off = 16; off > 0; off >>= 1) acc[i] += __shfl_down(acc[i], off, 32);
  }
  if (lane == 0) {
#pragma unroll
    for (int i = 0; i < 5; ++i) red[wave][i] = acc[i];
  }
  __syncthreads();

  if (threadIdx.x < 5) {
    double s = 0.0;
#pragma unroll
    for (int w = 0; w < WAVES; ++w) s += (double)red[w][threadIdx.x];
    atomicAdd(&ws[threadIdx.x], s);  // global_atomic_add_f64
  }
}

__global__ void finalize_kernel(const double* __restrict__ ws,
                                float* __restrict__ out, double inv_n) {
  const int i = threadIdx.x;
  if (i < 5) out[i] = (float)(ws[i] * inv_n);
}

extern "C" void kernel_launch(void* const* d_in, const int* in_sizes, int n_in,
                              void* d_out, int out_size, void* d_ws, size_t ws_size,
                              hipStream_t stream) {
  const float* pred = (const float*)d_in[0];
  const float* tgt  = (const float*)d_in[1];

  const long long totalf = (long long)in_sizes[0];   // N*7*7*30
  const long long cells  = totalf / CELL_F;          // N*49
  const int  nchunks     = (int)(cells / CHUNK_CELLS);
  const int  rem         = (int)(cells % CHUNK_CELLS);
  const long long N      = cells / 49;               // batch size (divisor)

  hipMemsetAsync(d_ws, 0, 5 * sizeof(double), stream);

  int blocks = 1568;  // 1568 * 8 waves = 12544 waves -> exactly 4 chunks/wave at N=32768
  if ((long long)blocks * WAVES > (long long)nchunks && nchunks > 0)
    blocks = (nchunks + WAVES - 1) / WAVES;
  if (blocks < 1) blocks = 1;

  yolo_loss_kernel<<<blocks, BLOCK, 0, stream>>>(
      pred, tgt, (double*)d_ws, nchunks, rem, (long long)nchunks * CHUNK_CELLS);

  finalize_kernel<<<1, 32, 0, stream>>>((const double*)d_ws, (float*)d_out,
                                        1.0 / (double)N);
}